// CombinedLoss_59803124630166
// MI455X (gfx1250) — compile-verified
//
#include <hip/hip_runtime.h>

// CombinedLoss: mse + adaptive-boundary pairwise ranking loss over B=8192.
// O(B^2) pairwise, pure VALU-bound (inputs = 64KB). Full-matrix symmetric sum
// (tri-sum = full/2), register-tiled columns, WMMA f32 16x16x4 wave reduction.
// v2: 4 independent accumulators (hide rcp/fma latency), uniform (scalar)
//     row-loop control instead of per-lane induction.

#define BETA_C  0.3f
#define GAMMA_C 0.1f
#define COLS    16
#define TPB     256

typedef __attribute__((ext_vector_type(2))) float v2f;
typedef __attribute__((ext_vector_type(8))) float v8f;

// Wave32 all-lane sum using V_WMMA_F32_16X16X4_F32 with a ones B-matrix.
// A (16x4 f32): lanes 0-15 -> A[m,0]=v0, A[m,1]=v1 ; lanes 16-31 -> A[m,2]=v0,
// A[m,3]=v1.  With v0=x, v1=0: D[m,n] = x_m + x_{m+16}.  Per-lane sum of the 8
// D vgprs covers M=0..7 (lanes<16) or M=8..15; one shfl_xor(16) completes the
// 32-lane total (valid in every lane).  EXEC must be all 1s at the call site.
__device__ __forceinline__ float wave_sum32_wmma(float x) {
  v2f a; a[0] = x;    a[1] = 0.0f;
  v2f b; b[0] = 1.0f; b[1] = 1.0f;
  v8f c = {0.f, 0.f, 0.f, 0.f, 0.f, 0.f, 0.f, 0.f};
  c = __builtin_amdgcn_wmma_f32_16x16x4_f32(
      /*neg_a=*/false, a, /*neg_b=*/false, b,
      /*c_mod=*/(short)0, c, /*reuse_a=*/false, /*reuse_b=*/false);
  float s = ((c[0] + c[1]) + (c[2] + c[3])) + ((c[4] + c[5]) + (c[6] + c[7]));
  s += __shfl_xor(s, 16, 32);
  return s;
}

// Each block owns COLS columns (register-cached, uniform scalar loads) and
// sweeps all B rows (coalesced). Writes one rank partial + one mse partial.
__global__ __launch_bounds__(TPB) void pair_partials_kernel(
    const float* __restrict__ pred, const float* __restrict__ target,
    float* __restrict__ ws, int B, int nwg) {
  const int tid = threadIdx.x;
  const int wg  = blockIdx.x;
  const int colBase = wg * COLS;

  // Uniform (per-block) column tile -> compiler scalarizes to SMEM loads.
  float tc[COLS], pc[COLS];
#pragma unroll
  for (int u = 0; u < COLS; ++u) {
    tc[u] = target[colBase + u];
    pc[u] = pred[colBase + u];
  }

  // 4 independent accumulator chains to hide v_rcp (TRANS) and fmac latency.
  float acc[4] = {0.0f, 0.0f, 0.0f, 0.0f};

  const int rows = B / TPB;          // uniform trip count -> SALU loop control
  for (int r = 0; r < rows; ++r) {
    const int i = r * TPB + tid;
    const float ti = target[i];
    const float pi = pred[i];
#pragma unroll
    for (int u = 0; u < COLS; ++u) {
      float d   = ti - tc[u];
      float ad  = fabsf(d);
      float q   = pi - pc[u];
      float qd  = q * d;                                     // q*sgn(d)*|d|
      float sq  = (d > 0.0f) ? q : ((d < 0.0f) ? -q : 0.0f); // q*sgn(d)
      float cde = fmaf(GAMMA_C, ad, 1.0f);                   // 1 + g*|d| > 0
      // max(0, b*|d|/c - sq) == max(0, b*|d| - sq - g*qd) * rcp(c)
      float num = fmaf(BETA_C, ad, -fmaf(GAMMA_C, qd, sq));
      acc[u & 3] = fmaf(fmaxf(num, 0.0f), __builtin_amdgcn_rcpf(cde), acc[u & 3]);
    }
  }
  const float accs = (acc[0] + acc[1]) + (acc[2] + acc[3]);

  __shared__ float lds[TPB / 32];
  const int lane = tid & 31;
  const int wv   = tid >> 5;

  // Rank partial: WMMA wave reduce -> fixed-order 8-wave sum (deterministic).
  float wsum = wave_sum32_wmma(accs);
  if (lane == 0) lds[wv] = wsum;
  __syncthreads();
  if (tid == 0) {
    float s = 0.0f;
#pragma unroll
    for (int w = 0; w < TPB / 32; ++w) s += lds[w];
    ws[wg] = s;
  }
  __syncthreads();

  // MSE partial: element g handled exactly once across the grid.
  const int g = wg * TPB + tid;
  float e = 0.0f;
  if (g < B) { float dd = pred[g] - target[g]; e = dd * dd; }
  float msum = wave_sum32_wmma(e);
  if (lane == 0) lds[wv] = msum;
  __syncthreads();
  if (tid == 0) {
    float s = 0.0f;
#pragma unroll
    for (int w = 0; w < TPB / 32; ++w) s += lds[w];
    ws[nwg + wg] = s;
  }
}

// Single-block deterministic final reduction + normalization.
__global__ __launch_bounds__(TPB) void finalize_kernel(
    const float* __restrict__ ws, float* __restrict__ out, int B, int nwg) {
  const int tid = threadIdx.x;
  float r = 0.0f, m = 0.0f;
  for (int idx = tid; idx < nwg; idx += TPB) {
    r += ws[idx];
    m += ws[nwg + idx];
  }
  __shared__ float lr[TPB / 32], lm[TPB / 32];
  float rw = wave_sum32_wmma(r);
  float mw = wave_sum32_wmma(m);
  const int lane = tid & 31;
  const int wv   = tid >> 5;
  if (lane == 0) { lr[wv] = rw; lm[wv] = mw; }
  __syncthreads();
  if (tid == 0) {
    float rs = 0.0f, ms = 0.0f;
#pragma unroll
    for (int w = 0; w < TPB / 32; ++w) { rs += lr[w]; ms += lm[w]; }
    float mse  = ms / (float)B;
    // full-matrix sum (incl. zero diagonal, both (i,j)&(j,i)) = 2 * tri-sum;
    // rank = full / (B*(B-1)).
    float rank = rs / ((float)B * (float)(B - 1));
    out[0] = mse + rank;  // combined
    out[1] = mse;
    out[2] = rank;
  }
}

extern "C" void kernel_launch(void* const* d_in, const int* in_sizes, int n_in,
                              void* d_out, int out_size, void* d_ws, size_t ws_size,
                              hipStream_t stream) {
  const float* pred   = (const float*)d_in[0];
  const float* target = (const float*)d_in[1];
  float* out = (float*)d_out;
  float* ws  = (float*)d_ws;   // [0,nwg): rank partials, [nwg,2*nwg): mse partials

  const int B   = in_sizes[0];     // 8192
  const int nwg = B / COLS;        // 512 blocks

  pair_partials_kernel<<<nwg, TPB, 0, stream>>>(pred, target, ws, B, nwg);
  finalize_kernel<<<1, TPB, 0, stream>>>(ws, out, B, nwg);
}